// PyGRGCNEdge_80324478369827
// MI455X (gfx1250) — compile-verified
//
#include <hip/hip_runtime.h>
#include <cmath>

#define D 128
#define R 8
#define L_LAYERS 2
#define BN_EPS 1e-5f

typedef __attribute__((ext_vector_type(2))) float v2f;
typedef __attribute__((ext_vector_type(8))) float v8f;

// ---------------------------------------------------------------------------
// 1) per-(relation,dst) edge counts
// ---------------------------------------------------------------------------
__global__ void rgcn_count_kernel(const int* __restrict__ dst,
                                  const int* __restrict__ etype,
                                  float* __restrict__ cnt,
                                  int n_nodes, int n_edges) {
  int e = blockIdx.x * blockDim.x + threadIdx.x;
  if (e >= n_edges) return;
  int r = etype[e];
  int d = dst[e];
  atomicAdd(&cnt[(size_t)r * n_nodes + d], 1.0f);
}

// ---------------------------------------------------------------------------
// 2) m = relu(h[src] + xe); scatter-add into sums[(r*N + dst)*D + k]
//    32 float4 chunks per edge (D/4 = 32)
// ---------------------------------------------------------------------------
__global__ void rgcn_scatter_kernel(const float* __restrict__ h,
                                    const float* __restrict__ xe,
                                    const int* __restrict__ src,
                                    const int* __restrict__ dst,
                                    const int* __restrict__ etype,
                                    float* __restrict__ sums,
                                    int n_nodes, int n_edges) {
  int idx = blockIdx.x * blockDim.x + threadIdx.x;
  int e = idx >> 5;          // edge
  int q = idx & 31;          // float4 chunk within the D=128 feature row
  if (e >= n_edges) return;
  int s  = src[e];
  int dn = dst[e];
  int r  = etype[e];
  float4 hv = ((const float4*)(h  + (size_t)s * D))[q];
  float4 xv = ((const float4*)(xe + (size_t)e * D))[q];
  float4 m;
  m.x = fmaxf(hv.x + xv.x, 0.0f);
  m.y = fmaxf(hv.y + xv.y, 0.0f);
  m.z = fmaxf(hv.z + xv.z, 0.0f);
  m.w = fmaxf(hv.w + xv.w, 0.0f);
  float* base = sums + ((size_t)r * n_nodes + dn) * D + (size_t)q * 4;
  atomicAdd(base + 0, m.x);
  atomicAdd(base + 1, m.y);
  atomicAdd(base + 2, m.z);
  atomicAdd(base + 3, m.w);
}

// ---------------------------------------------------------------------------
// 3) out[n,o] = sum_r mean_r[n,:] @ W_r[:,o]  +  h[n,:] @ root[:,o]  + bias[o]
//    One block per 16-node tile; 8 waves, each owning one 16-wide output
//    column tile.  fp32 WMMA 16x16x4.
//
//    fp32 WMMA fragment layouts (CDNA5 ISA 7.12.2), hi = lane>>4, mn = lane&15:
//      A (16x4):  a[j]  = A[mn][k0 + 2*hi + j]          (j = VGPR 0/1)
//      B (4x16):  b[j]  = B[k0 + 2*hi + j][mn]
//      C (16x16): c[v]  = C[v + 8*hi][mn]               (v = VGPR 0..7)
// ---------------------------------------------------------------------------
__global__ void __launch_bounds__(256)
rgcn_gemm_wmma(const float* __restrict__ sums,
               const float* __restrict__ cnt,
               const float* __restrict__ hin,
               const float* __restrict__ w,      // [R][D][D]  (k-major rows)
               const float* __restrict__ root,   // [D][D]
               const float* __restrict__ bias,   // [D]
               float* __restrict__ out,
               int n_nodes) {
  // Mean tile staged K-major: As[r][k][m] = sums[r, node0+m, k] / max(cnt,1)
  // 8*128*16*4 bytes = 64 KB exactly.  Read path (lanes sweep m) is
  // bank-conflict free; each element is reused by all 8 waves.
  __shared__ float As[R][D][16];

  const int node0 = blockIdx.x * 16;

  for (int i = threadIdx.x; i < R * 16 * D; i += 256) {
    int k = i & (D - 1);
    int m = (i >> 7) & 15;
    int r = i >> 11;
    int node = node0 + m;
    float v = 0.0f;
    if (node < n_nodes) {
      float c = cnt[(size_t)r * n_nodes + node];
      v = sums[((size_t)r * n_nodes + node) * D + k] / fmaxf(c, 1.0f);
    }
    As[r][k][m] = v;
  }
  __syncthreads();

  const int lane = threadIdx.x & 31;
  const int wv   = threadIdx.x >> 5;   // wave -> output column tile (0..7)
  const int hi   = lane >> 4;          // half-wave selector
  const int mn   = lane & 15;          // A row index == B/C column index
  const int ocol = wv * 16 + mn;

  v8f acc = {};

  // 8 relation GEMMs: A = mean_r tile (LDS), B = W_r (global, L2-resident)
  for (int r = 0; r < R; ++r) {
    const float* wr = w + (size_t)r * D * D;
#pragma unroll 8
    for (int k0 = 0; k0 < D; k0 += 4) {
      int ka = k0 + 2 * hi;
      v2f a, b;
      a.x = As[r][ka][mn];
      a.y = As[r][ka + 1][mn];
      b.x = wr[(size_t)ka * D + ocol];
      b.y = wr[(size_t)(ka + 1) * D + ocol];
      acc = __builtin_amdgcn_wmma_f32_16x16x4_f32(
          false, a, false, b, (short)0, acc, false, false);
    }
  }

  // Self/root term: A = h tile (global), B = root
  {
    int arow = node0 + mn;
    const float* hrow = hin + (size_t)(arow < n_nodes ? arow : 0) * D;
#pragma unroll 8
    for (int k0 = 0; k0 < D; k0 += 4) {
      int ka = k0 + 2 * hi;
      v2f a, b;
      a.x = hrow[ka];
      a.y = hrow[ka + 1];
      b.x = root[(size_t)ka * D + ocol];
      b.y = root[(size_t)(ka + 1) * D + ocol];
      acc = __builtin_amdgcn_wmma_f32_16x16x4_f32(
          false, a, false, b, (short)0, acc, false, false);
    }
  }

  float bv = bias[ocol];
#pragma unroll
  for (int v = 0; v < 8; ++v) {
    int row = node0 + v + 8 * hi;
    if (row < n_nodes)
      out[(size_t)row * D + ocol] = acc[v] + bv;
  }
}

// ---------------------------------------------------------------------------
// 4) BatchNorm stats: per-column sum / sumsq (block = 128 threads = 1 col each)
// ---------------------------------------------------------------------------
#define BN_ROWS 256
__global__ void bn_stats_kernel(const float* __restrict__ hb,
                                float* __restrict__ stats,
                                int n_nodes) {
  int col  = threadIdx.x;                 // 0..127
  int row0 = blockIdx.x * BN_ROWS;
  int lim  = n_nodes - row0;
  if (lim > BN_ROWS) lim = BN_ROWS;
  float s = 0.0f, s2 = 0.0f;
  for (int i = 0; i < lim; ++i) {
    float v = hb[(size_t)(row0 + i) * D + col];
    s  += v;
    s2 += v * v;
  }
  atomicAdd(&stats[col],     s);
  atomicAdd(&stats[D + col], s2);
}

// ---------------------------------------------------------------------------
// 5) BN apply (+ optional ReLU)
// ---------------------------------------------------------------------------
__global__ void bn_apply_kernel(const float* __restrict__ hb,
                                const float* __restrict__ stats,
                                const float* __restrict__ gamma,
                                const float* __restrict__ beta,
                                float* __restrict__ hout,
                                int n_nodes, int relu_flag) {
  int idx = blockIdx.x * blockDim.x + threadIdx.x;
  if (idx >= n_nodes * D) return;
  int col = idx & (D - 1);
  float inv_n = 1.0f / (float)n_nodes;
  float mu  = stats[col] * inv_n;
  float var = stats[D + col] * inv_n - mu * mu;   // biased variance
  float v = (hb[idx] - mu) * rsqrtf(var + BN_EPS) * gamma[col] + beta[col];
  if (relu_flag) v = fmaxf(v, 0.0f);
  hout[idx] = v;
}

// ---------------------------------------------------------------------------
extern "C" void kernel_launch(void* const* d_in, const int* in_sizes, int n_in,
                              void* d_out, int out_size, void* d_ws, size_t ws_size,
                              hipStream_t stream) {
  const float* x      = (const float*)d_in[0];   // [N, D]
  const float* xe     = (const float*)d_in[1];   // [E, D]
  const int*   ei     = (const int*)d_in[2];     // [2, E]
  const int*   et     = (const int*)d_in[3];     // [E]
  const float* weight = (const float*)d_in[4];   // [L, R, D, D]
  const float* root   = (const float*)d_in[5];   // [L, D, D]
  const float* bias   = (const float*)d_in[6];   // [L, D]
  const float* gamma  = (const float*)d_in[7];   // [L, D]
  const float* beta   = (const float*)d_in[8];   // [L, D]
  float* out = (float*)d_out;

  const int nN = in_sizes[0] / D;   // 50000
  const int nE = in_sizes[3];       // 500000
  const int* src = ei;
  const int* dst = ei + nE;

  // workspace carve-up (~258 MB)
  float* sums = (float*)d_ws;
  size_t sums_e = (size_t)R * nN * D;
  float* cntb  = sums + sums_e;
  float* hb    = cntb + (size_t)R * nN;
  float* h1    = hb   + (size_t)nN * D;
  float* stats = h1   + (size_t)nN * D;

  const long long scatter_threads = (long long)nE * 32;
  const int scatter_blocks = (int)((scatter_threads + 255) / 256);
  const long long elems    = (long long)nN * D;
  const int apply_blocks   = (int)((elems + 255) / 256);   // fixed: 25000, not 0

  for (int l = 0; l < L_LAYERS; ++l) {
    const float* hin  = (l == 0) ? x  : h1;
    float*       hout = (l == 0) ? h1 : out;

    hipMemsetAsync(sums,  0, sums_e * sizeof(float),            stream);
    hipMemsetAsync(cntb,  0, (size_t)R * nN * sizeof(float),    stream);
    hipMemsetAsync(stats, 0, 2 * D * sizeof(float),             stream);

    rgcn_count_kernel<<<(nE + 255) / 256, 256, 0, stream>>>(dst, et, cntb, nN, nE);
    rgcn_scatter_kernel<<<scatter_blocks, 256, 0, stream>>>(hin, xe, src, dst, et,
                                                            sums, nN, nE);
    rgcn_gemm_wmma<<<(nN + 15) / 16, 256, 0, stream>>>(
        sums, cntb, hin,
        weight + (size_t)l * R * D * D,
        root   + (size_t)l * D * D,
        bias   + (size_t)l * D,
        hb, nN);
    bn_stats_kernel<<<(nN + BN_ROWS - 1) / BN_ROWS, D, 0, stream>>>(hb, stats, nN);
    bn_apply_kernel<<<apply_blocks, 256, 0, stream>>>(
        hb, stats, gamma + (size_t)l * D, beta + (size_t)l * D,
        hout, nN, (l != L_LAYERS - 1) ? 1 : 0);
  }
}